// GCN_34935263986003
// MI455X (gfx1250) — compile-verified
//
#include <hip/hip_runtime.h>
#include <hip/hip_bf16.h>

typedef float v2f __attribute__((ext_vector_type(2)));
typedef float v8f __attribute__((ext_vector_type(8)));

#ifndef GCN_THREADS
#define GCN_THREADS 256
#endif

// ---------------------------------------------------------------------------
// Degree / normalization kernels
// ---------------------------------------------------------------------------
__global__ void k_deg_init(float* __restrict__ deg, int n) {
  int i = blockIdx.x * blockDim.x + threadIdx.x;
  if (i < n) deg[i] = 1.0f;  // self-loop contributes 1
}

__global__ void k_deg_count(const int* __restrict__ dst, float* __restrict__ deg, int e) {
  int i = blockIdx.x * blockDim.x + threadIdx.x;
  if (i < e) atomicAdd(&deg[dst[i]], 1.0f);
}

__global__ void k_dinv(float* __restrict__ deg, int n) {
  int i = blockIdx.x * blockDim.x + threadIdx.x;
  if (i < n) deg[i] = rsqrtf(deg[i]);  // deg >= 1 always (self-loop)
}

// ---------------------------------------------------------------------------
// fp32 WMMA GEMM:  out[M x NO] = A[M x K] * W[K x NO]
// One wave32 per 16-row tile; the wave accumulates all NT = NO/16 column
// tiles so the A panel is fetched once per wave.
//
// V_WMMA_F32_16X16X4_F32 VGPR striping (CDNA5 ISA 7.12.2):
//   A (16x4): lanes 0-15 -> M=lane,   VGPRs = K{0,1}; lanes 16-31 -> K{2,3}
//   B (4x16): lanes 0-15 -> N=lane,   VGPRs = K{0,1}; lanes 16-31 -> K{2,3}
//   C/D (16x16): VGPR r: lanes 0-15 -> M=r, N=lane; lanes 16-31 -> M=8+r
// ---------------------------------------------------------------------------
template <int K, int NO, int NT>
__global__ void k_gemm_wmma(const float* __restrict__ A, const float* __restrict__ W,
                            float* __restrict__ out, int mtiles) {
  const int wave = (int)((blockIdx.x * blockDim.x + threadIdx.x) >> 5);
  const int lane = (int)(threadIdx.x & 31);
  if (wave >= mtiles) return;  // uniform per-wave: surviving waves keep EXEC all-1

  const int hi   = lane >> 4;        // 0: lanes 0-15, 1: lanes 16-31
  const int lo   = lane & 15;
  const int kb   = hi << 1;          // K sub-offset 0 or 2
  const int row  = wave * 16 + lo;   // A row for this lane (M = lo)

  v8f acc[NT];
#pragma unroll
  for (int t = 0; t < NT; ++t) acc[t] = (v8f){0.f, 0.f, 0.f, 0.f, 0.f, 0.f, 0.f, 0.f};

  const float* __restrict__ arow = A + (size_t)row * K;

  for (int k0 = 0; k0 < K; k0 += 4) {
    // A fragment: two consecutive K values per lane -> single b64 load
    v2f a = *(const v2f*)(arow + k0 + kb);
#pragma unroll
    for (int t = 0; t < NT; ++t) {
      const int n = t * 16 + lo;
      v2f b;
      b.x = W[(size_t)(k0 + kb + 0) * NO + n];
      b.y = W[(size_t)(k0 + kb + 1) * NO + n];
      acc[t] = __builtin_amdgcn_wmma_f32_16x16x4_f32(
          /*neg_a=*/false, a, /*neg_b=*/false, b,
          /*c_mod=*/(short)0, acc[t], /*reuse_a=*/false, /*reuse_b=*/false);
    }
  }

  const int mbase = wave * 16 + (hi << 3);  // lanes 0-15 -> +0, 16-31 -> +8
#pragma unroll
  for (int t = 0; t < NT; ++t) {
#pragma unroll
    for (int r = 0; r < 8; ++r) {
      out[(size_t)(mbase + r) * NO + t * 16 + lo] = acc[t][r];
    }
  }
}

// ---------------------------------------------------------------------------
// acc[i][f] = h[i][f] * dinv[i]^2   (self-loop term initializes accumulator)
// ---------------------------------------------------------------------------
template <int D>
__global__ void k_selfloop_init(const float* __restrict__ h, const float* __restrict__ dinv,
                                float* __restrict__ acc, int n) {
  const unsigned QP = D / 4;
  unsigned i = blockIdx.x * blockDim.x + threadIdx.x;
  if (i >= (unsigned)n * QP) return;
  unsigned node = i / QP;
  unsigned f4   = (i % QP) * 4;
  float s = dinv[node];
  s = s * s;
  float4 v = *(const float4*)(h + (size_t)node * D + f4);
  v.x *= s; v.y *= s; v.z *= s; v.w *= s;
  *(float4*)(acc + (size_t)node * D + f4) = v;
}

// ---------------------------------------------------------------------------
// scatter-add over edges: acc[dst] += h[src] * dinv[src]*dinv[dst]
// float4 gather, 4x global_atomic_add_f32 (L2-resident accumulator)
// ---------------------------------------------------------------------------
template <int D>
__global__ void k_edge_scatter(const int* __restrict__ src, const int* __restrict__ dst,
                               const float* __restrict__ dinv, const float* __restrict__ h,
                               float* __restrict__ acc, int e) {
  const unsigned QP = D / 4;
  unsigned tid = blockIdx.x * blockDim.x + threadIdx.x;
  if (tid >= (unsigned)e * QP) return;
  unsigned eid = tid / QP;
  unsigned f4  = (tid % QP) * 4;
  int s = src[eid];
  int d = dst[eid];
  float nrm = dinv[s] * dinv[d];
  float4 v = *(const float4*)(h + (size_t)s * D + f4);
  float* o = acc + (size_t)d * D + f4;
  atomicAdd(o + 0, v.x * nrm);
  atomicAdd(o + 1, v.y * nrm);
  atomicAdd(o + 2, v.z * nrm);
  atomicAdd(o + 3, v.w * nrm);
}

// out[i*D+f] = relu(acc[i*D+f] + b[f])
template <int D>
__global__ void k_bias_relu(const float* __restrict__ acc, const float* __restrict__ b,
                            float* __restrict__ out, unsigned total) {
  unsigned i = blockIdx.x * blockDim.x + threadIdx.x;
  if (i >= total) return;
  out[i] = fmaxf(acc[i] + b[i & (D - 1)], 0.0f);
}

// out[i*D+f] += b[f]
template <int D>
__global__ void k_bias_add(float* __restrict__ out, const float* __restrict__ b, unsigned total) {
  unsigned i = blockIdx.x * blockDim.x + threadIdx.x;
  if (i >= total) return;
  out[i] += b[i & (D - 1)];
}

// ---------------------------------------------------------------------------
// Host-side launch sequence (graph-capture safe: only async launches)
// ---------------------------------------------------------------------------
extern "C" void kernel_launch(void* const* d_in, const int* in_sizes, int n_in,
                              void* d_out, int out_size, void* d_ws, size_t ws_size,
                              hipStream_t stream) {
  (void)n_in; (void)out_size; (void)ws_size;

  const float* x    = (const float*)d_in[0];  // [N,128]
  const int*   ei   = (const int*)d_in[1];    // [2,E] row0=src row1=dst
  const float* W1   = (const float*)d_in[2];  // [128,128]
  const float* b1   = (const float*)d_in[3];  // [128]
  const float* W2   = (const float*)d_in[4];  // [128,64]
  const float* b2   = (const float*)d_in[5];  // [64]
  float*       out  = (float*)d_out;          // [N,64]

  const int N = in_sizes[0] / 128;
  const int E = in_sizes[1] / 2;
  const int IN_D = 128, HID_D = 128, OUT_D = 64;

  const int* src = ei;
  const int* dst = ei + E;

  // Workspace layout (floats): dinv[N] | bufA[N*128] | bufB[N*128]
  float* dinv = (float*)d_ws;
  float* bufA = dinv + N;                 // h1, then h2 (relu output)
  float* bufB = bufA + (size_t)N * 128;   // acc1, then g2

  const int T = GCN_THREADS;
  auto blk = [&](long long work) { return (unsigned)((work + T - 1) / T); };

  // 1) symmetric normalization: dinv = rsqrt(deg + selfloop)
  k_deg_init<<<blk(N), T, 0, stream>>>(dinv, N);
  k_deg_count<<<blk(E), T, 0, stream>>>(dst, dinv, E);
  k_dinv<<<blk(N), T, 0, stream>>>(dinv, N);

  // 2) layer 1 dense transform: h1 = x @ W1   (fp32 WMMA)
  const int mtiles = N / 16;  // N = 100000 -> 6250 tiles
  {
    const int wavesPerBlk = T / 32;
    unsigned g = (unsigned)((mtiles + wavesPerBlk - 1) / wavesPerBlk);
    k_gemm_wmma<IN_D, HID_D, HID_D / 16><<<g, T, 0, stream>>>(x, W1, bufA, mtiles);
  }

  // 3) layer 1 propagate: acc1 = selfloop + edge scatter; h2 = relu(acc1 + b1)
  k_selfloop_init<HID_D><<<blk((long long)N * (HID_D / 4)), T, 0, stream>>>(bufA, dinv, bufB, N);
  k_edge_scatter<HID_D><<<blk((long long)E * (HID_D / 4)), T, 0, stream>>>(src, dst, dinv, bufA, bufB, E);
  k_bias_relu<HID_D><<<blk((long long)N * HID_D), T, 0, stream>>>(bufB, b1, bufA, (unsigned)N * HID_D);

  // 4) layer 2 dense transform: g2 = h2 @ W2   (fp32 WMMA)
  {
    const int wavesPerBlk = T / 32;
    unsigned g = (unsigned)((mtiles + wavesPerBlk - 1) / wavesPerBlk);
    k_gemm_wmma<HID_D, OUT_D, OUT_D / 16><<<g, T, 0, stream>>>(bufA, W2, bufB, mtiles);
  }

  // 5) layer 2 propagate directly into d_out, then bias
  k_selfloop_init<OUT_D><<<blk((long long)N * (OUT_D / 4)), T, 0, stream>>>(bufB, dinv, out, N);
  k_edge_scatter<OUT_D><<<blk((long long)E * (OUT_D / 4)), T, 0, stream>>>(src, dst, dinv, bufB, out, E);
  k_bias_add<OUT_D><<<blk((long long)N * OUT_D), T, 0, stream>>>(out, b2, (unsigned)N * OUT_D);
}